// MLP_33028298506661
// MI455X (gfx1250) — compile-verified
//
#include <hip/hip_runtime.h>
#include <hip/hip_bf16.h>
#include <math.h>

#define N_NODES    50000
#define N_EDGES    800000
#define NUM_GRAPHS 256
#define H_DIM      64
#define HID        256
#define BN_EPS     1e-5f

#define GEMM_WAVES 8          // waves per workgroup (one 16-row tile each)
#define KCHUNK     128        // K staged per LDS round (128*64*4B = 32 KB)

typedef float v2f  __attribute__((ext_vector_type(2)));
typedef float v8f  __attribute__((ext_vector_type(8)));
typedef unsigned int u32x4 __attribute__((ext_vector_type(4)));
typedef int   i32x4 __attribute__((ext_vector_type(4)));
typedef int   i32x8 __attribute__((ext_vector_type(8)));

__device__ __forceinline__ v8f wmma_f32(v2f a, v2f b, v8f c) {
    return __builtin_amdgcn_wmma_f32_16x16x4_f32(
        /*neg_a=*/false, a, /*neg_b=*/false, b,
        /*c_mod=*/(short)0, c, /*reuse_a=*/false, /*reuse_b=*/false);
}

// ---------------------------------------------------------------------------
// C[M,Nc] = A[M,K] @ W[K,Nc] (+ optional bias), f32 WMMA 16x16x4.
// Workgroup = 8 waves; each wave computes one 16x64 output strip (4 accums).
// The 64-col weight strip W[kc:kc+kl, ntb*64:+64] is staged into LDS once per
// workgroup per K-chunk via the Tensor Data Mover (tensor_load_to_lds,
// TENSORcnt), then all 8 waves feed their WMMA B-fragments from LDS.
// A memory-clobber asm makes the DMA's LDS writes visible to the optimizer
// (the builtin has no pointer arg, so without it the smem loads fold to undef).
// Requires M % 16 == 0, K % 4 == 0, Nc % 64 == 0.
// A frag (16x4 f32): lane holds row M = lane&15; VGPR i holds K = i + 2*(lane>=16).
// B frag (4x16 f32): lane holds col N = lane&15; VGPR i holds K = i + 2*(lane>=16).
// C/D frag: VGPR i -> row i + 8*(lane>=16), col = lane&15.
// ---------------------------------------------------------------------------
__global__ void gemm_wmma_f32(const float* __restrict__ A,
                              const float* __restrict__ W,
                              const float* __restrict__ bias,
                              float* __restrict__ C,
                              int M, int K, int Nc) {
    __shared__ float smem[KCHUNK * 64];

    const int wave = threadIdx.x >> 5;           // 0..7
    const int lane = threadIdx.x & 31;
    const int hi   = lane >> 4;                  // 0 or 1
    const int lo   = lane & 15;
    const int ntb  = blockIdx.y;                 // 64-col strip
    const int mt   = blockIdx.x * GEMM_WAVES + wave;
    const bool active = (mt * 16) < M;

    v8f acc[4] = {};
    const float* arow = A + (size_t)(mt * 16 + lo) * K + 2 * hi;

    for (int kc = 0; kc < K; kc += KCHUNK) {
        const int kl = (K - kc) < KCHUNK ? (K - kc) : KCHUNK;

#if __has_builtin(__builtin_amdgcn_tensor_load_to_lds) && defined(__gfx1250__)
        if (threadIdx.x < 32) {                  // wave 0 issues one TDM DMA
            // D# group 0: count=1 | lds_addr | global_addr | type=2
            const unsigned long long gaddr =
                (unsigned long long)(const char*)(W + (size_t)kc * Nc + ntb * 64);
            // generic LDS-aperture address: byte offset within LDS = addr[31:0]
            const unsigned lds_off = (unsigned)(unsigned long long)(const char*)&smem[0];
            u32x4 g0;
            g0[0] = 1u;                                       // count = 1 (valid)
            g0[1] = lds_off;                                  // lds_addr (bytes)
            g0[2] = (unsigned)(gaddr & 0xffffffffu);          // global_addr[31:0]
            g0[3] = (unsigned)(gaddr >> 32) | (2u << 30);     // [56:32] | type=2
            // D# group 1: data_size=4B; tensor 64 x kl; tile 64 x kl; stride0 = Nc
            i32x8 g1;
            g1[0] = (int)(2u << 16);                          // data_size = 4 bytes
            g1[1] = (int)(64u << 16);                         // tensor_dim0[15:0]
            g1[2] = (int)(((unsigned)kl & 0xffffu) << 16);    // tensor_dim1[15:0]
            g1[3] = (int)(64u << 16);                         // tile_dim0 = 64
            g1[4] = (int)((unsigned)kl & 0xffffu);            // tile_dim1 = kl
            g1[5] = (int)(unsigned)Nc;                        // tensor_dim0_stride
            g1[6] = 0;
            g1[7] = 0;
            i32x4 gz4 = {};                                   // groups 2/3: 2-D tile
            i32x8 gz8 = {};                                   // trailing group (unused)
            __builtin_amdgcn_tensor_load_to_lds(g0, g1, gz4, gz4, gz8, 0);
            __builtin_amdgcn_s_wait_tensorcnt(0);
        }
#else
        // fallback: cooperative global -> LDS copy
        for (int idx = threadIdx.x; idx < kl * 64; idx += blockDim.x) {
            const int r = idx >> 6, c = idx & 63;
            smem[idx] = W[(size_t)(kc + r) * Nc + ntb * 64 + c];
        }
#endif
        __syncthreads();                         // strip visible to all 8 waves
        // The TDM wrote smem behind the compiler's back; this clobber gives the
        // LDS loads below a defining memory access so they are not folded away.
        asm volatile("" ::: "memory");

        if (active) {
            for (int k0 = 0; k0 < kl; k0 += 4) {
                v2f a;
                a.x = arow[kc + k0];
                a.y = arow[kc + k0 + 1];
                const float* b0 = &smem[(k0 + 2 * hi) * 64 + lo];
#pragma unroll
                for (int t = 0; t < 4; ++t) {
                    v2f b;
                    b.x = b0[t * 16];            // row k0+2*hi
                    b.y = b0[64 + t * 16];       // row k0+2*hi+1
                    acc[t] = wmma_f32(a, b, acc[t]);
                }
            }
        }
        __syncthreads();                         // before next chunk overwrites
    }

    if (active) {
        const int colBase = ntb * 64 + lo;
#pragma unroll
        for (int t = 0; t < 4; ++t) {
            const int col = colBase + t * 16;
            const float bb = bias ? bias[col] : 0.0f;
#pragma unroll
            for (int i = 0; i < 8; ++i) {
                const int row = mt * 16 + i + 8 * hi;
                C[(size_t)row * Nc + col] = acc[t][i] + bb;
            }
        }
    }
}

// in-degree counts (float, reused by all three layers)
__global__ void edge_count(const int* __restrict__ dst, float* __restrict__ cnt) {
    int e = blockIdx.x * blockDim.x + threadIdx.x;
    if (e >= N_EDGES) return;
    atomicAdd(&cnt[dst[e]], 1.0f);
}

// agg[dst[e], f] += y[src[e], f]  (64 features, one thread per (edge, feature))
__global__ void edge_scatter(const int* __restrict__ src, const int* __restrict__ dst,
                             const float* __restrict__ y, float* __restrict__ agg) {
    unsigned idx = blockIdx.x * blockDim.x + threadIdx.x;   // < E*64 = 51.2M
    unsigned e = idx >> 6;
    int f = idx & 63;
    if (e >= (unsigned)N_EDGES) return;
    atomicAdd(&agg[(size_t)dst[e] * H_DIM + f], y[(size_t)src[e] * H_DIM + f]);
}

// out[i,f] = agg[i,f] / max(cnt[i],1) + bl[f] + yr[i,f]
__global__ void combine_kernel(const float* __restrict__ agg, const float* __restrict__ cnt,
                               const float* __restrict__ bl, const float* __restrict__ yr,
                               float* __restrict__ out) {
    unsigned idx = blockIdx.x * blockDim.x + threadIdx.x;   // < N*64
    unsigned i = idx >> 6;
    int f = idx & 63;
    if (i >= (unsigned)N_NODES) return;
    out[idx] = agg[idx] / fmaxf(cnt[i], 1.0f) + bl[f] + yr[idx];
}

// global mean pool: scatter phase
__global__ void pool_scatter(const int* __restrict__ batch, const float* __restrict__ h,
                             float* __restrict__ pool, float* __restrict__ gcnt) {
    unsigned idx = blockIdx.x * blockDim.x + threadIdx.x;   // < N*64
    unsigned i = idx >> 6;
    int f = idx & 63;
    if (i >= (unsigned)N_NODES) return;
    int g = batch[i];
    atomicAdd(&pool[(size_t)g * H_DIM + f], h[idx]);
    if (f == 0) atomicAdd(&gcnt[g], 1.0f);
}

__global__ void pool_div(float* __restrict__ pool, const float* __restrict__ gcnt) {
    int idx = blockIdx.x * blockDim.x + threadIdx.x;        // 256*64
    int g = idx >> 6;
    pool[idx] /= fmaxf(gcnt[g], 1.0f);
}

// training-mode BN (biased var over 256 rows) + tanh; one block per column.
__global__ void bn_tanh_kernel(float* __restrict__ z, const float* __restrict__ g,
                               const float* __restrict__ be, int cols) {
    __shared__ float red[256];
    const int j = blockIdx.x;            // column
    const int i = threadIdx.x;           // row (blockDim == 256 == NUM_GRAPHS)
    float v = z[(size_t)i * cols + j];
    red[i] = v;
    __syncthreads();
    for (int s = 128; s > 0; s >>= 1) { if (i < s) red[i] += red[i + s]; __syncthreads(); }
    const float mean = red[0] * (1.0f / 256.0f);
    __syncthreads();
    const float d = v - mean;
    red[i] = d * d;
    __syncthreads();
    for (int s = 128; s > 0; s >>= 1) { if (i < s) red[i] += red[i + s]; __syncthreads(); }
    const float var = red[0] * (1.0f / 256.0f);
    z[(size_t)i * cols + j] = tanhf(d * rsqrtf(var + BN_EPS) * g[j] + be[j]);
}

// final 256x64 @ 64x10 + bias (Nc=10 not a multiple of 16 -> scalar tail kernel)
__global__ void lin4_kernel(const float* __restrict__ z3, const float* __restrict__ w,
                            const float* __restrict__ b, float* __restrict__ out) {
    int idx = blockIdx.x * blockDim.x + threadIdx.x;        // 2560
    if (idx >= NUM_GRAPHS * 10) return;
    int r = idx / 10, c = idx % 10;
    float s = b[c];
#pragma unroll
    for (int k = 0; k < H_DIM; ++k) s += z3[r * H_DIM + k] * w[k * 10 + c];
    out[idx] = s;
}

static inline dim3 gemm_grid(int M, int Nc) {
    return dim3((M / 16 + GEMM_WAVES - 1) / GEMM_WAVES, Nc / 64);
}

extern "C" void kernel_launch(void* const* d_in, const int* in_sizes, int n_in,
                              void* d_out, int out_size, void* d_ws, size_t ws_size,
                              hipStream_t stream) {
    (void)in_sizes; (void)n_in; (void)out_size; (void)ws_size;

    const float* x      = (const float*)d_in[0];
    const int*   ei     = (const int*)d_in[1];
    const int*   src    = ei;
    const int*   dst    = ei + N_EDGES;
    const int*   batch  = (const int*)d_in[2];
    const float* W1l = (const float*)d_in[3],  *b1l = (const float*)d_in[4],  *W1r = (const float*)d_in[5];
    const float* W2l = (const float*)d_in[6],  *b2l = (const float*)d_in[7],  *W2r = (const float*)d_in[8];
    const float* W3l = (const float*)d_in[9],  *b3l = (const float*)d_in[10], *W3r = (const float*)d_in[11];
    const float* lin1_w = (const float*)d_in[12], *lin1_b = (const float*)d_in[13];
    const float* g1 = (const float*)d_in[14], *be1 = (const float*)d_in[15];
    const float* lin2_w = (const float*)d_in[16], *lin2_b = (const float*)d_in[17];
    const float* g2 = (const float*)d_in[18], *be2 = (const float*)d_in[19];
    const float* lin3_w = (const float*)d_in[20], *lin3_b = (const float*)d_in[21];
    const float* g3 = (const float*)d_in[22], *be3 = (const float*)d_in[23];
    const float* lin4_w = (const float*)d_in[24], *lin4_b = (const float*)d_in[25];
    float* out = (float*)d_out;

    // workspace carve-out (~52 MB of f32)
    float* ws   = (float*)d_ws;
    float* cnt  = ws;  ws += N_NODES;
    float* yl   = ws;  ws += (size_t)N_NODES * H_DIM;
    float* yr   = ws;  ws += (size_t)N_NODES * H_DIM;
    float* agg  = ws;  ws += (size_t)N_NODES * H_DIM;
    float* hbf  = ws;  ws += (size_t)N_NODES * H_DIM;
    float* pool = ws;  ws += NUM_GRAPHS * H_DIM;
    float* gcnt = ws;  ws += NUM_GRAPHS;
    float* z1   = ws;  ws += NUM_GRAPHS * HID;
    float* z2   = ws;  ws += NUM_GRAPHS * (HID / 2);
    float* z3   = ws;  ws += NUM_GRAPHS * (HID / 4);

    const int EF_BLOCKS = (N_EDGES * H_DIM + 255) / 256;    // edge*feature grid
    const int NF_BLOCKS = (N_NODES * H_DIM + 255) / 256;    // node*feature grid
    const int GEMM_BLK  = GEMM_WAVES * 32;                  // 256 threads

    // in-degree counts (shared across layers)
    (void)hipMemsetAsync(cnt, 0, N_NODES * sizeof(float), stream);
    edge_count<<<(N_EDGES + 255) / 256, 256, 0, stream>>>(dst, cnt);

    // one SAGE layer: project-then-aggregate (mean commutes with right-matmul)
    auto sage_layer = [&](const float* hin, int K, const float* Wl, const float* bl,
                          const float* Wr) {
        dim3 grid = gemm_grid(N_NODES, H_DIM);
        gemm_wmma_f32<<<grid, GEMM_BLK, 0, stream>>>(hin, Wl, nullptr, yl, N_NODES, K, H_DIM);
        gemm_wmma_f32<<<grid, GEMM_BLK, 0, stream>>>(hin, Wr, nullptr, yr, N_NODES, K, H_DIM);
        (void)hipMemsetAsync(agg, 0, (size_t)N_NODES * H_DIM * sizeof(float), stream);
        edge_scatter<<<EF_BLOCKS, 256, 0, stream>>>(src, dst, yl, agg);
        combine_kernel<<<NF_BLOCKS, 256, 0, stream>>>(agg, cnt, bl, yr, hbf);
    };

    sage_layer(x,   128, W1l, b1l, W1r);
    sage_layer(hbf,  64, W2l, b2l, W2r);
    sage_layer(hbf,  64, W3l, b3l, W3r);

    // global mean pool (pool and gcnt are contiguous in the carve-out)
    (void)hipMemsetAsync(pool, 0, (NUM_GRAPHS * H_DIM + NUM_GRAPHS) * sizeof(float), stream);
    pool_scatter<<<NF_BLOCKS, 256, 0, stream>>>(batch, hbf, pool, gcnt);
    pool_div<<<(NUM_GRAPHS * H_DIM) / 256, 256, 0, stream>>>(pool, gcnt);

    // MLP head (all WMMA GEMMs with fused bias); Nc ∈ {256,128,64}, all % 64 == 0
    gemm_wmma_f32<<<gemm_grid(NUM_GRAPHS, HID), GEMM_BLK, 0, stream>>>(
        pool, lin1_w, lin1_b, z1, NUM_GRAPHS, H_DIM, HID);
    bn_tanh_kernel<<<HID, 256, 0, stream>>>(z1, g1, be1, HID);

    gemm_wmma_f32<<<gemm_grid(NUM_GRAPHS, HID / 2), GEMM_BLK, 0, stream>>>(
        z1, lin2_w, lin2_b, z2, NUM_GRAPHS, HID, HID / 2);
    bn_tanh_kernel<<<HID / 2, 256, 0, stream>>>(z2, g2, be2, HID / 2);

    gemm_wmma_f32<<<gemm_grid(NUM_GRAPHS, HID / 4), GEMM_BLK, 0, stream>>>(
        z2, lin3_w, lin3_b, z3, NUM_GRAPHS, HID / 2, HID / 4);
    bn_tanh_kernel<<<HID / 4, 256, 0, stream>>>(z3, g3, be3, HID / 4);

    lin4_kernel<<<(NUM_GRAPHS * 10 + 255) / 256, 256, 0, stream>>>(z3, lin4_w, lin4_b, out);
}